// FittedGaussianPoseLoss_74053826118244
// MI455X (gfx1250) — compile-verified
//
#include <hip/hip_runtime.h>
#include <hip/hip_bf16.h>

typedef __attribute__((ext_vector_type(2))) float v2f;
typedef __attribute__((ext_vector_type(4))) float f4;
typedef __attribute__((ext_vector_type(8))) float v8f;

#define BLOCK 256
#define NPTS 20

// CDNA5 async global->LDS copy (ASYNCcnt-tracked, no VGPR round-trip).
// GV form: vdst = LDS byte address (VGPR), vaddr = 64-bit global address.
__device__ __forceinline__ void async_copy_b128(unsigned int lds_addr, const void* gaddr) {
    asm volatile("global_load_async_to_lds_b128 %0, %1, off"
                 :: "v"(lds_addr), "v"(gaddr) : "memory");
}
__device__ __forceinline__ void wait_asynccnt0() {
    asm volatile("s_wait_asynccnt 0x0" ::: "memory");
}

// Principal square root of symmetric PSD 2x2 [[a,b],[b,c]], times 2
// (folds the "2*M @ points" factor from the reference).
__device__ __forceinline__ void sym_sqrt_x2(float sx, float sy, float th,
                                            float& m00, float& m01, float& m11) {
    float a = sx * sx;
    float c = sy * sy;
    float b = th * sx * sy;
    float det = fmaxf(fmaf(a, c, -b * b), 0.0f);
    float s = sqrtf(det);
    float t = sqrtf(fmaxf(a + c + 2.0f * s, 0.0f));
    float inv = (t > 0.0f) ? (2.0f / t) : 0.0f;
    m00 = (a + s) * inv;
    m01 = b * inv;
    m11 = (c + s) * inv;
}

__global__ __launch_bounds__(BLOCK)
void gauss_loss_partial(const float* __restrict__ pred,
                        const float* __restrict__ gth,
                        float* __restrict__ partial,
                        int B) {
    __shared__ __align__(16) float sp[BLOCK * 14];
    __shared__ __align__(16) float sg[BLOCK * 14];
    __shared__ float spx[NPTS];
    __shared__ float spy[NPTS];
    __shared__ float wsum[BLOCK / 32];

    const int tid = threadIdx.x;
    const size_t rowbase = (size_t)blockIdx.x * BLOCK;
    const size_t totalf = (size_t)B * 14;

    // Stage 256 rows x 14 floats of both tensors into LDS.
    // Fast path: CDNA5 async global->LDS b128 copies (overlap with sincos below).
    if ((rowbase + BLOCK) * 14 <= totalf) {
        const f4* p4 = (const f4*)(pred + rowbase * 14);
        const f4* g4 = (const f4*)(gth + rowbase * 14);
        // Low 32 bits of the generic LDS address == LDS byte offset (aperture map).
        unsigned int sp_lds = (unsigned int)(uintptr_t)(void*)sp;
        unsigned int sg_lds = (unsigned int)(uintptr_t)(void*)sg;
        for (int i = tid; i < (BLOCK * 14) / 4; i += BLOCK) {
            async_copy_b128(sp_lds + 16u * (unsigned)i, p4 + i);
            async_copy_b128(sg_lds + 16u * (unsigned)i, g4 + i);
        }
    } else {
        for (int i = tid; i < BLOCK * 14; i += BLOCK) {
            size_t g = rowbase * 14 + (size_t)i;
            sp[i] = (g < totalf) ? pred[g] : 0.0f;
            sg[i] = (g < totalf) ? gth[g] : 0.0f;
        }
    }

    // Circle sample points: theta = linspace(0, 2pi, 20) (endpoint included).
    // Computed while the async copies are in flight.
    if (tid < NPTS) {
        float th = 6.283185307179586f * ((float)tid / 19.0f);
        float s, c;
        __sincosf(th, &s, &c);
        spx[tid] = 0.75f * c;
        spy[tid] = 0.75f * s;
    }

    wait_asynccnt0();   // each wave drains its own async copies
    __syncthreads();    // then cross-wave visibility of LDS

    const float* rp = sp + tid * 14;
    const float* rg = sg + tid * 14;

    float thread_sum = 0.0f;
    #pragma unroll
    for (int setk = 0; setk < 2; ++setk) {
        const int o = setk * 7;
        float pxo = rp[o + 1], pyo = rp[o + 2];
        float psx = rp[o + 3], psy = rp[o + 4], pth = rp[o + 5];
        float gxo = rg[o + 1], gyo = rg[o + 2];
        float gsx = rg[o + 3], gsy = rg[o + 4], gtv = rg[o + 5];

        float pm00, pm01, pm11, gm00, gm01, gm11;
        sym_sqrt_x2(psx, psy, pth, pm00, pm01, pm11);
        sym_sqrt_x2(gsx, gsy, gtv, gm00, gm01, gm11);

        int np = 0, ng = 0;
        float dsum = 0.0f;
        #pragma unroll
        for (int p = 0; p < NPTS; ++p) {
            float px = spx[p], py = spy[p];
            // u = x-row (checked vs 128), v = y-row (checked vs 96)
            float pu = fmaf(pm00, px, fmaf(pm01, py, pxo));
            float pv = fmaf(pm01, px, fmaf(pm11, py, pyo));
            float gu = fmaf(gm00, px, fmaf(gm01, py, gxo));
            float gv = fmaf(gm01, px, fmaf(gm11, py, gyo));
            np += (pv > 0.0f) && (pv < 96.0f) && (pu > 0.0f) && (pu < 128.0f);
            ng += (gv > 0.0f) && (gv < 96.0f) && (gu > 0.0f) && (gu < 128.0f);
            float du = pu - gu;
            float dv = pv - gv;
            dsum = fmaf(du, du, fmaf(dv, dv, dsum));
        }
        float f;
        if (setk == 0) {
            // gth_fact = (ng > n/4); pred_fact = (np - ng < -n/2); both {0,1}
            f = ((ng > 5) && (np - ng < -10)) ? 1.0f : 0.0f;
        } else {
            // gth_fact = (ng > n/2); pred_fact in {1,101}; squared -> {1,10201}
            f = (ng > 10) ? ((np - ng < -10) ? 10201.0f : 1.0f) : 0.0f;
        }
        thread_sum = fmaf(f, dsum, thread_sum);
    }

    // Zero tail rows without disturbing EXEC (WMMA needs all lanes active).
    if (rowbase + (size_t)tid >= (size_t)B) thread_sum = 0.0f;

    // Wave32 reduction via V_WMMA_F32_16X16X4_F32 with B = ones:
    // row sums => D[m,*] = p[m] + p[m+16] (fp32 matrix-unit add tree).
    v2f a;
    a[0] = thread_sum;
    a[1] = 0.0f;
    v2f bones;
    bones[0] = 1.0f;
    bones[1] = 1.0f;
    v8f cacc = {};
    v8f d = __builtin_amdgcn_wmma_f32_16x16x4_f32(
        /*neg_a=*/false, a, /*neg_b=*/false, bones,
        /*c_mod=*/(short)0, cacc, /*reuse_a=*/false, /*reuse_b=*/false);
    float s = d[0] + d[1] + d[2] + d[3] + d[4] + d[5] + d[6] + d[7];
    // lanes 0-15 hold rows 0-7, lanes 16-31 hold rows 8-15: combine halves.
    s += __shfl_xor(s, 16, 32);

    if ((tid & 31) == 0) wsum[tid >> 5] = s;
    __syncthreads();
    if (tid == 0) {
        float t = 0.0f;
        #pragma unroll
        for (int i = 0; i < BLOCK / 32; ++i) t += wsum[i];
        partial[blockIdx.x] = t;
    }
}

__global__ __launch_bounds__(BLOCK)
void gauss_loss_final(const float* __restrict__ partial, int nparts,
                      float* __restrict__ out, double Bd) {
    __shared__ double sd[BLOCK];
    double acc = 0.0;
    for (int i = threadIdx.x; i < nparts; i += BLOCK) acc += (double)partial[i];
    sd[threadIdx.x] = acc;
    __syncthreads();
    for (int off = BLOCK / 2; off > 0; off >>= 1) {
        if (threadIdx.x < off) sd[threadIdx.x] += sd[threadIdx.x + off];
        __syncthreads();
    }
    if (threadIdx.x == 0) {
        double Bn = Bd * 20.0;        // B * n
        double numel = Bd * 40.0;     // B * n * 2
        out[0] = (float)((sd[0] / Bn) / numel);
    }
}

extern "C" void kernel_launch(void* const* d_in, const int* in_sizes, int n_in,
                              void* d_out, int out_size, void* d_ws, size_t ws_size,
                              hipStream_t stream) {
    const float* pred = (const float*)d_in[0];
    const float* gth  = (const float*)d_in[1];
    float* out = (float*)d_out;
    float* partials = (float*)d_ws;

    const int B = in_sizes[0] / 14;              // 524288
    const int nblocks = (B + BLOCK - 1) / BLOCK; // 2048 -> 8 KB of d_ws

    gauss_loss_partial<<<nblocks, BLOCK, 0, stream>>>(pred, gth, partials, B);
    gauss_loss_final<<<1, BLOCK, 0, stream>>>(partials, nblocks, out, (double)B);
}